// TemporalAttentionCausal_64458869179035
// MI455X (gfx1250) — compile-verified
//
#include <hip/hip_runtime.h>

typedef __attribute__((ext_vector_type(2))) float v2f;
typedef __attribute__((ext_vector_type(8))) float v8f;

#define T_DIM 1024
#define H_DIM 256

// One workgroup (256 threads = 8 waves) per (b,l) row.
// Phase A: scores, global max, exp, cum_exp scan -> LDS e[], inv[].
// Phase C: WMMA-tiled inclusive cumsum of e[t]*h[t,:] along T (16-t tiles,
//          L16 lower-triangular ones matrix, carry via C accumulator).
//          Each wave owns 32 channels = two interleaved 16-col WMMA chains.
__global__ __launch_bounds__(256, 1)
void temporal_attn_fused(const float* __restrict__ h,
                         const float* __restrict__ w,
                         const float* __restrict__ b,
                         float* __restrict__ out) {
  __shared__ float sW[H_DIM];
  __shared__ float sE[T_DIM];
  __shared__ float sInv[T_DIM];
  __shared__ float sRed[256];

  const int tid = threadIdx.x;
  const float* __restrict__ hrow = h + (size_t)blockIdx.x * T_DIM * H_DIM;
  float* __restrict__ orow = out + (size_t)blockIdx.x * T_DIM * H_DIM;

  sW[tid] = w[tid];
  const float bias = b[0];
  __syncthreads();

  // ---------------- Phase A: scores for t = 4*tid + i ----------------
  float sv[4];
  float lmax = -3.402823466e38f;
  {
    const float4* wp = (const float4*)sW;
    #pragma unroll
    for (int i = 0; i < 4; ++i) {
      const int t = 4 * tid + i;
      const float4* hp = (const float4*)(hrow + (size_t)t * H_DIM);
      float acc = 0.f;
      #pragma unroll 16
      for (int jj = 0; jj < H_DIM / 4; ++jj) {
        float4 hv = hp[jj];
        float4 wv = wp[jj];
        acc = fmaf(hv.x, wv.x, acc);
        acc = fmaf(hv.y, wv.y, acc);
        acc = fmaf(hv.z, wv.z, acc);
        acc = fmaf(hv.w, wv.w, acc);
      }
      sv[i] = acc + bias;
      lmax = fmaxf(lmax, sv[i]);
    }
  }

  // ---- global max over T (block tree reduction) ----
  sRed[tid] = lmax;
  __syncthreads();
  #pragma unroll
  for (int off = 128; off > 0; off >>= 1) {
    if (tid < off) sRed[tid] = fmaxf(sRed[tid], sRed[tid + off]);
    __syncthreads();
  }
  const float smax = sRed[0];
  __syncthreads();

  // ---- exp + local inclusive prefix over this thread's 4 t's ----
  float ev[4], pf[4];
  {
    float run = 0.f;
    #pragma unroll
    for (int i = 0; i < 4; ++i) {
      ev[i] = __expf(sv[i] - smax);
      run += ev[i];
      pf[i] = run;
    }
    sRed[tid] = run;  // block sum of 4
  }
  __syncthreads();

  // ---- Hillis-Steele inclusive scan of 256 block sums ----
  #pragma unroll
  for (int off = 1; off < 256; off <<= 1) {
    float v = sRed[tid];
    float addv = (tid >= off) ? sRed[tid - off] : 0.f;
    __syncthreads();
    sRed[tid] = v + addv;
    __syncthreads();
  }
  const float excl = (tid > 0) ? sRed[tid - 1] : 0.f;

  #pragma unroll
  for (int i = 0; i < 4; ++i) {
    const int t = 4 * tid + i;
    sE[t] = ev[i];
    const float cum = excl + pf[i];
    sInv[t] = 1.0f / (cum + 1e-12f);
  }
  __syncthreads();

  // ---------------- Phase C: WMMA tiled scan along T ----------------
  const int lane   = tid & 31;
  const int waveId = tid >> 5;
  const int nl     = lane & 15;  // column (N) / row (M) index within tile
  const int khalf  = lane >> 4;  // which K-half this lane holds

  // Constant lower-triangular A (16x4 f32 slices): A[m][k] = (k <= m)
  // f32 A layout: comp c of lane holds K = 4*ks + 2*khalf + c, M = lane&15.
  v2f Amat[4];
  #pragma unroll
  for (int ks = 0; ks < 4; ++ks) {
    const int k0 = 4 * ks + 2 * khalf;
    Amat[ks][0] = (k0     <= nl) ? 1.0f : 0.0f;
    Amat[ks][1] = (k0 + 1 <= nl) ? 1.0f : 0.0f;
  }

  // This wave owns 32 channels: two 16-col tiles at n0 and n0+16,
  // processed as two independent interleaved WMMA scan chains.
  const int n0 = waveId * 32;
  v8f acc0 = {0.f, 0.f, 0.f, 0.f, 0.f, 0.f, 0.f, 0.f};
  v8f acc1 = {0.f, 0.f, 0.f, 0.f, 0.f, 0.f, 0.f, 0.f};

  for (int tile = 0; tile < T_DIM / 16; ++tile) {
    const int t0 = tile * 16;

    // B slices for both column tiles: B[k][n] = e[t0+k] * h[t0+k, n0(+16)+n]
    // f32 B layout: comp c of lane holds K = 4*ks + 2*khalf + c, N = lane&15.
    v2f Bm0[4], Bm1[4];
    #pragma unroll
    for (int ks = 0; ks < 4; ++ks) {
      const int ka = t0 + 4 * ks + 2 * khalf;
      const float ea = sE[ka];
      const float eb = sE[ka + 1];
      const size_t ra = (size_t)ka * H_DIM + n0 + nl;
      const size_t rb = (size_t)(ka + 1) * H_DIM + n0 + nl;
      Bm0[ks][0] = ea * hrow[ra];
      Bm0[ks][1] = eb * hrow[rb];
      Bm1[ks][0] = ea * hrow[ra + 16];
      Bm1[ks][1] = eb * hrow[rb + 16];
    }

    // D = L * W + carry  (two independent chains of 4 K-slices)
    #pragma unroll
    for (int ks = 0; ks < 4; ++ks) {
      acc0 = __builtin_amdgcn_wmma_f32_16x16x4_f32(
          false, Amat[ks], false, Bm0[ks], (short)0, acc0, false, false);
      acc1 = __builtin_amdgcn_wmma_f32_16x16x4_f32(
          false, Amat[ks], false, Bm1[ks], (short)0, acc1, false, false);
    }

    // out[t,n] = h[t,n] + D[t,n] * inv[t]   (non-temporal stores: out is
    // never re-read; keep L2 for h which IS re-read)
    // D layout: VGPR r -> row t0+r (lanes 0-15) / t0+8+r (lanes 16-31).
    #pragma unroll
    for (int r = 0; r < 8; ++r) {
      const int t = t0 + r + 8 * khalf;
      const float iv = sInv[t];
      const size_t idx = (size_t)t * H_DIM + n0 + nl;
      __builtin_nontemporal_store(hrow[idx] + acc0[r] * iv, &orow[idx]);
      __builtin_nontemporal_store(hrow[idx + 16] + acc1[r] * iv, &orow[idx + 16]);
    }

    // carry for next tile = D[15][n]  (VGPR7, lanes 16-31, N = lane-16)
    const float carry0 = __shfl(acc0[7], 16 + nl, 32);
    const float carry1 = __shfl(acc1[7], 16 + nl, 32);
    #pragma unroll
    for (int r = 0; r < 8; ++r) {
      acc0[r] = carry0;
      acc1[r] = carry1;
    }
  }
}

extern "C" void kernel_launch(void* const* d_in, const int* in_sizes, int n_in,
                              void* d_out, int out_size, void* d_ws, size_t ws_size,
                              hipStream_t stream) {
  const float* h = (const float*)d_in[0];   // [B,L,T,H] f32
  const float* w = (const float*)d_in[1];   // [1,H]     f32
  const float* b = (const float*)d_in[2];   // [1]       f32
  float* out = (float*)d_out;               // [B,L,T,H] f32
  (void)d_ws; (void)ws_size; (void)n_in;

  const int rows = in_sizes[0] / (T_DIM * H_DIM);  // B*L = 96
  temporal_attn_fused<<<dim3(rows), dim3(256), 0, stream>>>(h, w, b, out);
}